// GraphSAGE_89601607729381
// MI455X (gfx1250) — compile-verified
//
#include <hip/hip_runtime.h>

#define N_NODES 50000
#define N_EDGES 800000
#define D_FEAT  128

typedef __attribute__((ext_vector_type(2))) float v2f;
typedef __attribute__((ext_vector_type(8))) float v8f;

// ---------------------------------------------------------------------------
// Zero a contiguous float region (grid-stride).
// ---------------------------------------------------------------------------
__global__ void zero_kernel(float* __restrict__ p, size_t n) {
  size_t i = (size_t)blockIdx.x * blockDim.x + threadIdx.x;
  size_t stride = (size_t)gridDim.x * blockDim.x;
  for (; i < n; i += stride) p[i] = 0.0f;
}

// ---------------------------------------------------------------------------
// deg[dst[e]] += 1.0f  (f32 atomic add, L2-resident)
// ---------------------------------------------------------------------------
__global__ void degree_kernel(const int* __restrict__ dst,
                              float* __restrict__ deg) {
  int e = blockIdx.x * blockDim.x + threadIdx.x;
  if (e < N_EDGES) {
    __hip_atomic_fetch_add(&deg[dst[e]], 1.0f, __ATOMIC_RELAXED,
                           __HIP_MEMORY_SCOPE_AGENT);
  }
}

// ---------------------------------------------------------------------------
// Edge-parallel scatter-add: one wave32 per edge, float4 per lane.
// agg[dst[e]][:] += feat[src[e]][:]   (128 floats / edge)
// ---------------------------------------------------------------------------
__global__ void scatter_kernel(const float* __restrict__ feat,
                               const int* __restrict__ src,
                               const int* __restrict__ dst,
                               float* __restrict__ agg) {
  int e = (int)((blockIdx.x * blockDim.x + threadIdx.x) >> 5);
  int lane = threadIdx.x & 31;
  if (e >= N_EDGES) return;
  int s = src[e];
  int d = dst[e];
  float4 v = *(const float4*)(feat + (size_t)s * D_FEAT + lane * 4);
  float* p = agg + (size_t)d * D_FEAT + lane * 4;
  __hip_atomic_fetch_add(p + 0, v.x, __ATOMIC_RELAXED, __HIP_MEMORY_SCOPE_AGENT);
  __hip_atomic_fetch_add(p + 1, v.y, __ATOMIC_RELAXED, __HIP_MEMORY_SCOPE_AGENT);
  __hip_atomic_fetch_add(p + 2, v.z, __ATOMIC_RELAXED, __HIP_MEMORY_SCOPE_AGENT);
  __hip_atomic_fetch_add(p + 3, v.w, __ATOMIC_RELAXED, __HIP_MEMORY_SCOPE_AGENT);
}

// ---------------------------------------------------------------------------
// Fused SAGEConv output GEMM (fp32 WMMA, exact precision):
//   out = (agg/deg) @ Wl + xself @ Wr + bias   [+ ReLU]
//
// Block: 256 threads = 8 wave32s; computes 64 rows x DOUT cols.
// Wave tiling: tiles t = wave + 8*i; rt = t/CT (row tile), ct = t%CT (col).
// Weights staged through LDS in 16xDOUT chunks (8KB max).
// WMMA fp32 fragment layouts per ISA 7.12.2:
//   A 16x4 : lane l -> row l%16, K = 2*(l/16) + {0,1}
//   B  4x16: VGPR j, lane l -> K = j + 2*(l/16), col = l%16
//   C 16x16: VGPR j, lane l -> row = j + 8*(l/16), col = l%16
//
// Per 16-K chunk, all 4 A-fragment b64 loads are issued before the WMMA
// sequence so the compiler can clause them and overlap latency with WMMA.
// ---------------------------------------------------------------------------
template <int DOUT, bool RELU>
__global__ __launch_bounds__(256) void sage_gemm_kernel(
    const float* __restrict__ xself,  // [N][128] self features
    const float* __restrict__ agg,    // [N][128] neighbor sums
    const float* __restrict__ deg,    // [N] degrees (float)
    const float* __restrict__ Wl,     // [128][DOUT] mean path
    const float* __restrict__ Wr,     // [128][DOUT] self path
    const float* __restrict__ bias,   // [DOUT]
    float* __restrict__ out) {        // [N][DOUT]
  constexpr int CT = DOUT / 16;        // column tiles
  constexpr int OWN = (4 * CT) / 8;    // tiles owned per wave
  __shared__ float sW[16 * DOUT];

  const int tid = threadIdx.x;
  const int wave = tid >> 5;
  const int lane = tid & 31;
  const int lhalf = lane >> 4;  // 0|1
  const int lmod = lane & 15;
  const int rowbase = blockIdx.x * 64;

  int rtA[OWN], ctA[OWN];
  v8f acc[OWN];
#pragma unroll
  for (int i = 0; i < OWN; ++i) {
    int t = wave + 8 * i;
    rtA[i] = t / CT;
    ctA[i] = t % CT;
    float bv = bias[ctA[i] * 16 + lmod];
#pragma unroll
    for (int j = 0; j < 8; ++j) acc[i][j] = bv;
  }

#pragma unroll
  for (int pass = 0; pass < 2; ++pass) {
    const float* __restrict__ W = pass ? Wl : Wr;
    const float* __restrict__ A = pass ? agg : xself;
    for (int k0 = 0; k0 < D_FEAT; k0 += 16) {
      __syncthreads();
      // cooperative copy of a 16 x DOUT weight chunk into LDS
      for (int idx = tid * 4; idx < 16 * DOUT; idx += 256 * 4) {
        *(float4*)(sW + idx) = *(const float4*)(W + (size_t)k0 * DOUT + idx);
      }
      __syncthreads();
#pragma unroll
      for (int i = 0; i < OWN; ++i) {
        int r = rowbase + rtA[i] * 16 + lmod;
        r = (r < N_NODES) ? r : (N_NODES - 1);  // clamp: keep EXEC all-1s
        float sc = 1.0f;
        if (pass) sc = 1.0f / fmaxf(deg[r], 1.0f);
        const int c = ctA[i] * 16 + lmod;
        const float* __restrict__ Ar =
            A + (size_t)r * D_FEAT + k0 + 2 * lhalf;
        // issue all 4 independent A-fragment loads up front (clauseable)
        float2 av[4];
#pragma unroll
        for (int q = 0; q < 4; ++q) {
          av[q] = *(const float2*)(Ar + q * 4);
        }
#pragma unroll
        for (int q = 0; q < 4; ++q) {
          const int kb = q * 4 + 2 * lhalf;
          v2f a;
          a.x = av[q].x * sc;
          a.y = av[q].y * sc;
          v2f b;
          b.x = sW[kb * DOUT + c];
          b.y = sW[(kb + 1) * DOUT + c];
          acc[i] = __builtin_amdgcn_wmma_f32_16x16x4_f32(
              false, a, false, b, (short)0, acc[i], false, false);
        }
      }
    }
  }

  // predicated store of C tiles
#pragma unroll
  for (int i = 0; i < OWN; ++i) {
    const int c = ctA[i] * 16 + lmod;
#pragma unroll
    for (int j = 0; j < 8; ++j) {
      int row = rowbase + rtA[i] * 16 + j + 8 * lhalf;
      if (row < N_NODES) {
        float v = acc[i][j];
        if (RELU) v = fmaxf(v, 0.0f);
        out[(size_t)row * DOUT + c] = v;
      }
    }
  }
}

// ---------------------------------------------------------------------------
// Host launcher. Inputs (setup_inputs order):
//   0:x[N,128] f32  1:edge_index[2,E] int  2:W1_l[128,128]  3:W1_r[128,128]
//   4:b1[128]       5:W2_l[128,64]    6:W2_r[128,64]        7:b2[64]
// Output: [N,64] f32
// ---------------------------------------------------------------------------
extern "C" void kernel_launch(void* const* d_in, const int* in_sizes, int n_in,
                              void* d_out, int out_size, void* d_ws,
                              size_t ws_size, hipStream_t stream) {
  const float* x   = (const float*)d_in[0];
  const int*   ei  = (const int*)d_in[1];
  const float* W1l = (const float*)d_in[2];
  const float* W1r = (const float*)d_in[3];
  const float* b1  = (const float*)d_in[4];
  const float* W2l = (const float*)d_in[5];
  const float* W2r = (const float*)d_in[6];
  const float* b2  = (const float*)d_in[7];
  float* out = (float*)d_out;

  const int* src = ei;            // edge_index[0]
  const int* dst = ei + N_EDGES;  // edge_index[1]

  // workspace layout (floats): deg | agg | agg2 | h
  float* ws = (float*)d_ws;
  size_t off = 0;
  float* deg  = ws + off; off += ((N_NODES + 255) / 256) * 256;
  float* agg  = ws + off; off += (size_t)N_NODES * D_FEAT;
  float* agg2 = ws + off; off += (size_t)N_NODES * D_FEAT;
  float* h    = ws + off; off += (size_t)N_NODES * D_FEAT;
  (void)ws_size;

  // zero deg + agg + agg2 (contiguous prefix; h is fully overwritten)
  size_t nzero = (size_t)(agg2 - deg) + (size_t)N_NODES * D_FEAT;
  zero_kernel<<<2048, 256, 0, stream>>>(deg, nzero);

  degree_kernel<<<(N_EDGES + 255) / 256, 256, 0, stream>>>(dst, deg);

  // layer 1: aggregate x, then h = relu(mean@W1l + x@W1r + b1)
  scatter_kernel<<<(N_EDGES * 32) / 256, 256, 0, stream>>>(x, src, dst, agg);
  sage_gemm_kernel<128, true>
      <<<(N_NODES + 63) / 64, 256, 0, stream>>>(x, agg, deg, W1l, W1r, b1, h);

  // layer 2: aggregate h, then out = mean@W2l + h@W2r + b2
  scatter_kernel<<<(N_EDGES * 32) / 256, 256, 0, stream>>>(h, src, dst, agg2);
  sage_gemm_kernel<64, false>
      <<<(N_NODES + 63) / 64, 256, 0, stream>>>(h, agg2, deg, W2l, W2r, b2, out);
}